// GCNNet_74861279969818
// MI455X (gfx1250) — compile-verified
//
#include <hip/hip_runtime.h>
#include <hip/hip_bf16.h>
#include <math.h>

typedef __attribute__((ext_vector_type(2))) float v2f;
typedef __attribute__((ext_vector_type(8))) float v8f;

#define WMMA_F32(a, b, c) \
  __builtin_amdgcn_wmma_f32_16x16x4_f32(false, (a), false, (b), (short)0, (c), false, false)

// ---------------------------------------------------------------- degree prep
__global__ __launch_bounds__(256) void k_init_deg(float* __restrict__ deg, int n) {
  int i = blockIdx.x * 256 + threadIdx.x;
  if (i < n) deg[i] = 1.0f;  // self-loop contributes 1
}

__global__ __launch_bounds__(256) void k_deg_edges(const int* __restrict__ col,
                                                   float* __restrict__ deg, int E) {
  int e = blockIdx.x * 256 + threadIdx.x;
  if (e < E) atomicAdd(&deg[col[e]], 1.0f);
}

__global__ __launch_bounds__(256) void k_dinv(float* __restrict__ deg, int n) {
  int i = blockIdx.x * 256 + threadIdx.x;
  if (i < n) deg[i] = rsqrtf(deg[i]);  // deg >= 1 always (self loops)
}

// ---------------------------------------------------------------- GEMM1: h1 = x @ W1 (WMMA f32 16x16x4)
// Each wave computes one 16-row tile across all 64 output columns (4 WMMA col-tiles).
// Epilogue also writes g1 = dinv^2 * h1 (self-loop term of the aggregation).
__global__ __launch_bounds__(256) void k_gemm1_wmma(
    const float* __restrict__ x, const float* __restrict__ W1,
    const float* __restrict__ dinv, float* __restrict__ h1,
    float* __restrict__ g1, int nTiles) {
  __shared__ float sW1[64 * 64];
  {
    const float4* src = (const float4*)W1;
    float4* dst = (float4*)sW1;
    for (int i = threadIdx.x; i < 64 * 64 / 4; i += 256) dst[i] = src[i];
  }
  __syncthreads();

  const int wave = threadIdx.x >> 5;
  const int lane = threadIdx.x & 31;
  const int tile = blockIdx.x * 8 + wave;
  if (tile >= nTiles) return;

  const int m16 = lane & 15;   // A row / B,C column within tile
  const int kHi = lane >> 4;   // selects K pair {0,1} vs {2,3}
  const float* xrow = x + (size_t)(tile * 16 + m16) * 64;

  v8f acc0 = {}, acc1 = {}, acc2 = {}, acc3 = {};
  for (int k0 = 0; k0 < 64; k0 += 4) {
    const int ka = k0 + 2 * kHi;
    v2f a;
    a.x = xrow[ka];
    a.y = xrow[ka + 1];
    const float* bA = sW1 + ka * 64 + m16;        // row ka
    const float* bB = sW1 + (ka + 1) * 64 + m16;  // row ka+1
    v2f b;
    b.x = bA[0];  b.y = bB[0];  acc0 = WMMA_F32(a, b, acc0);
    b.x = bA[16]; b.y = bB[16]; acc1 = WMMA_F32(a, b, acc1);
    b.x = bA[32]; b.y = bB[32]; acc2 = WMMA_F32(a, b, acc2);
    b.x = bA[48]; b.y = bB[48]; acc3 = WMMA_F32(a, b, acc3);
  }

#pragma unroll
  for (int j = 0; j < 8; ++j) {
    const int rm = tile * 16 + j + 8 * kHi;  // C/D layout: M = j + 8*(lane/16)
    const float dv = dinv[rm];
    const float d2 = dv * dv;
    const size_t cb = (size_t)rm * 64 + m16;
    h1[cb +  0] = acc0[j]; g1[cb +  0] = acc0[j] * d2;
    h1[cb + 16] = acc1[j]; g1[cb + 16] = acc1[j] * d2;
    h1[cb + 32] = acc2[j]; g1[cb + 32] = acc2[j] * d2;
    h1[cb + 48] = acc3[j]; g1[cb + 48] = acc3[j] * d2;
  }
}

// Scalar tail for rows not covered by full 16-row tiles (unused when n % 16 == 0).
__global__ __launch_bounds__(256) void k_gemm1_tail(
    const float* __restrict__ x, const float* __restrict__ W1,
    const float* __restrict__ dinv, float* __restrict__ h1,
    float* __restrict__ g1, int rowStart, int n) {
  int g = blockIdx.x * 256 + threadIdx.x;
  int r = rowStart + (g >> 6), c = g & 63;
  if (r >= n) return;
  float s = 0.f;
  for (int k = 0; k < 64; ++k) s += x[(size_t)r * 64 + k] * W1[k * 64 + c];
  float dv = dinv[r];
  h1[(size_t)r * 64 + c] = s;
  g1[(size_t)r * 64 + c] = s * dv * dv;
}

// ---------------------------------------------------------------- layer-1 edge scatter
// 16 lanes per edge, one float4 each => fully coalesced 256B gather per edge.
__global__ __launch_bounds__(256) void k_scatter1(
    const int* __restrict__ row, const int* __restrict__ col,
    const float* __restrict__ dinv, const float* __restrict__ h1,
    float* __restrict__ g1, int E) {
  int g = blockIdx.x * 256 + threadIdx.x;
  int e = g >> 4, part = g & 15;
  if (e >= E) return;
  const int r = row[e], c = col[e];
  const float w = dinv[r] * dinv[c];
  const float4 v = ((const float4*)(h1 + (size_t)r * 64))[part];
  float* dst = g1 + (size_t)c * 64 + part * 4;
  atomicAdd(dst + 0, v.x * w);
  atomicAdd(dst + 1, v.y * w);
  atomicAdd(dst + 2, v.z * w);
  atomicAdd(dst + 3, v.w * w);
}

// ---------------------------------------------------------------- GEMM2: h2 = relu(g1+b1) @ W2 (WMMA, B padded 2->16)
// Fuses bias+ReLU into A-fragment loads; epilogue writes h2 and out = b2 + dinv^2*h2.
__global__ __launch_bounds__(256) void k_gemm2_wmma(
    const float* __restrict__ g1, const float* __restrict__ b1,
    const float* __restrict__ W2, const float* __restrict__ b2,
    const float* __restrict__ dinv, float* __restrict__ h2,
    float* __restrict__ out, int nTiles) {
  __shared__ float sW2[64 * 2];
  __shared__ float sb1[64];
  if (threadIdx.x < 128) sW2[threadIdx.x] = W2[threadIdx.x];
  if (threadIdx.x < 64) sb1[threadIdx.x] = b1[threadIdx.x];
  __syncthreads();

  const int wave = threadIdx.x >> 5;
  const int lane = threadIdx.x & 31;
  const int tile = blockIdx.x * 8 + wave;
  if (tile >= nTiles) return;

  const int m16 = lane & 15;
  const int kHi = lane >> 4;
  const float* arow = g1 + (size_t)(tile * 16 + m16) * 64;

  v8f acc = {};
  for (int k0 = 0; k0 < 64; k0 += 4) {
    const int ka = k0 + 2 * kHi;
    v2f a, b;
    a.x = fmaxf(arow[ka] + sb1[ka], 0.f);
    a.y = fmaxf(arow[ka + 1] + sb1[ka + 1], 0.f);
    b.x = (m16 < 2) ? sW2[ka * 2 + m16] : 0.f;        // zero-pad cols 2..15
    b.y = (m16 < 2) ? sW2[(ka + 1) * 2 + m16] : 0.f;
    acc = WMMA_F32(a, b, acc);
  }

  if (m16 < 2) {
#pragma unroll
    for (int j = 0; j < 8; ++j) {
      const int rm = tile * 16 + j + 8 * kHi;
      const float v = acc[j];
      const float dv = dinv[rm];
      h2[(size_t)rm * 2 + m16] = v;
      out[(size_t)rm * 2 + m16] = b2[m16] + v * dv * dv;  // self-loop + bias init
    }
  }
}

__global__ __launch_bounds__(256) void k_gemm2_tail(
    const float* __restrict__ g1, const float* __restrict__ b1,
    const float* __restrict__ W2, const float* __restrict__ b2,
    const float* __restrict__ dinv, float* __restrict__ h2,
    float* __restrict__ out, int rowStart, int n) {
  int g = blockIdx.x * 256 + threadIdx.x;
  int r = rowStart + (g >> 1), c = g & 1;
  if (r >= n) return;
  float s = 0.f;
  for (int k = 0; k < 64; ++k)
    s += fmaxf(g1[(size_t)r * 64 + k] + b1[k], 0.f) * W2[k * 2 + c];
  float dv = dinv[r];
  h2[(size_t)r * 2 + c] = s;
  out[(size_t)r * 2 + c] = b2[c] + s * dv * dv;
}

// ---------------------------------------------------------------- layer-2 edge scatter (2 channels)
__global__ __launch_bounds__(256) void k_scatter2(
    const int* __restrict__ row, const int* __restrict__ col,
    const float* __restrict__ dinv, const float* __restrict__ h2,
    float* __restrict__ out, int E) {
  int e = blockIdx.x * 256 + threadIdx.x;
  if (e >= E) return;
  const int r = row[e], c = col[e];
  const float w = dinv[r] * dinv[c];
  const float2 v = *(const float2*)(h2 + (size_t)r * 2);
  atomicAdd(&out[(size_t)c * 2 + 0], v.x * w);
  atomicAdd(&out[(size_t)c * 2 + 1], v.y * w);
}

// ---------------------------------------------------------------- launch
extern "C" void kernel_launch(void* const* d_in, const int* in_sizes, int n_in,
                              void* d_out, int out_size, void* d_ws, size_t ws_size,
                              hipStream_t stream) {
  const float* x  = (const float*)d_in[0];
  const int*   ei = (const int*)d_in[1];
  const float* W1 = (const float*)d_in[2];
  const float* b1 = (const float*)d_in[3];
  const float* W2 = (const float*)d_in[4];
  const float* b2 = (const float*)d_in[5];
  float* out = (float*)d_out;

  const int n = in_sizes[0] / 64;   // 100000
  const int E = in_sizes[1] / 2;    // 1600000
  const int* row = ei;
  const int* col = ei + E;

  // workspace layout: dinv[n] | h1[n*64] | g1[n*64] | h2[n*2]
  float* ws = (float*)d_ws;
  float* dinv = ws;
  float* h1 = dinv + n;
  float* g1 = h1 + (size_t)n * 64;
  float* h2 = g1 + (size_t)n * 64;

  const int tiles = n / 16;
  const int rem = n - tiles * 16;
  const int nB = (n + 255) / 256;
  const int eB = (E + 255) / 256;

  k_init_deg<<<nB, 256, 0, stream>>>(dinv, n);
  k_deg_edges<<<eB, 256, 0, stream>>>(col, dinv, E);
  k_dinv<<<nB, 256, 0, stream>>>(dinv, n);

  if (tiles)
    k_gemm1_wmma<<<(tiles + 7) / 8, 256, 0, stream>>>(x, W1, dinv, h1, g1, tiles);
  if (rem)
    k_gemm1_tail<<<(rem * 64 + 255) / 256, 256, 0, stream>>>(x, W1, dinv, h1, g1, tiles * 16, n);

  k_scatter1<<<(E * 16 + 255) / 256, 256, 0, stream>>>(row, col, dinv, h1, g1, E);

  if (tiles)
    k_gemm2_wmma<<<(tiles + 7) / 8, 256, 0, stream>>>(g1, b1, W2, b2, dinv, h2, out, tiles);
  if (rem)
    k_gemm2_tail<<<(rem * 2 + 255) / 256, 256, 0, stream>>>(g1, b1, W2, b2, dinv, h2, out, tiles * 16, n);

  k_scatter2<<<eB, 256, 0, stream>>>(row, col, dinv, h2, out, E);
}